// ListwiseRankingLoss_11218454577738
// MI455X (gfx1250) — compile-verified
//
#include <hip/hip_runtime.h>
#include <math.h>

typedef float v2f __attribute__((ext_vector_type(2)));
typedef float v8f __attribute__((ext_vector_type(8)));

static constexpr int Bn = 1024;  // batch
static constexpr int An = 64;    // answers per row
static constexpr int Dn = 1024;  // embedding dim
#define EPSV 1e-8f

// One block (128 threads = 4 wave32) per batch row b.
// Phase 1: sim[b, 0..63] via v_wmma_f32_16x16x4_f32, B-matrix = q chunk
//          replicated across all 16 columns (every column accumulates the
//          same dot product -> C[m, *] = sim[m]).
// Phase 2: 64 threads: softmax/KL + pairwise rank counting for NDCG.
__global__ __launch_bounds__(128) void sim_rank_kernel(
    const float* __restrict__ qemb,     // [B][D]
    const float* __restrict__ ans,      // [B][A][D]
    const float* __restrict__ scores,   // [B][A]
    float* __restrict__ kl_out,         // [B]
    float* __restrict__ nd_out)         // [B]
{
  __shared__ float lds_q[Dn];
  __shared__ float lds_sim[An];
  __shared__ float lds_sc[An];
  __shared__ float lds_kl[An];
  __shared__ float lds_pd[An];
  __shared__ float lds_id[An];

  const int b    = blockIdx.x;
  const int tid  = threadIdx.x;
  const int lane = tid & 31;
  const int wave = tid >> 5;

  // Stage q[b][:] into LDS (128 threads x 8 dwords).
  {
    const float* qrow = qemb + (size_t)b * Dn;
#pragma unroll
    for (int i = 0; i < Dn / 128; ++i)
      lds_q[tid + i * 128] = qrow[tid + i * 128];
  }
  if (tid < An) lds_sc[tid] = scores[(size_t)b * An + tid];
  __syncthreads();

  // ---- Phase 1: WMMA dot products; wave w owns answers [16w, 16w+16). ----
  {
    const int m     = wave * 16 + (lane & 15);
    const int khalf = (lane >> 4) << 1;                 // 0 (lanes 0-15) or 2 (lanes 16-31)
    const float* arow = ans + ((size_t)b * An + m) * Dn + khalf;
    const float* qoff = lds_q + khalf;

    v8f c = {0.f, 0.f, 0.f, 0.f, 0.f, 0.f, 0.f, 0.f};
#pragma unroll 8
    for (int k = 0; k < Dn; k += 4) {
      // A-matrix 16x4 f32 layout: lane l -> row m=l%16, (K, K+1) with K = khalf.
      v2f av = __builtin_nontemporal_load((const v2f*)(arow + k));  // streaming, read-once
      // B-matrix 4x16 f32, replicated across N: lanes<16 need (q[k],q[k+1]),
      // lanes>=16 need (q[k+2],q[k+3]) -> same khalf offset. LDS broadcast read.
      v2f qv = *(const v2f*)(qoff + k);
      c = __builtin_amdgcn_wmma_f32_16x16x4_f32(
              /*neg_a=*/false, av, /*neg_b=*/false, qv,
              /*c_mod=*/(short)0, c, /*reuse_a=*/false, /*reuse_b=*/false);
    }
    // C layout: VGPR i = M=i (lanes 0-15) / M=i+8 (lanes 16-31); all N equal.
    if (lane == 0) {
#pragma unroll
      for (int i = 0; i < 8; ++i) lds_sim[wave * 16 + i] = c[i];
    }
    if (lane == 16) {
#pragma unroll
      for (int i = 0; i < 8; ++i) lds_sim[wave * 16 + 8 + i] = c[i];
    }
  }
  __syncthreads();

  // ---- Phase 2: softmax/KL + NDCG ranks (A^2 = 4096 broadcast LDS reads). ----
  if (tid < An) {
    const float st = lds_sim[tid];
    const float sc = lds_sc[tid];
    float msim = -INFINITY, msc = -INFINITY;
    for (int j = 0; j < An; ++j) {
      msim = fmaxf(msim, lds_sim[j]);
      msc  = fmaxf(msc,  lds_sc[j]);
    }
    float esim = 0.f, esc = 0.f;
    int rp = 0, ri = 0;  // rank = #{strictly larger} + #{equal with lower index}
    for (int j = 0; j < An; ++j) {
      const float sj = lds_sim[j];
      const float cj = lds_sc[j];
      esim += __expf(sj - msim);
      esc  += __expf(cj - msc);
      rp += (sj > st) || ((sj == st) && (j < tid));
      ri += (cj > sc) || ((cj == sc) && (j < tid));
    }
    const float p  = __expf(st - msim) / esim;          // softmax(sim)
    const float tg = __expf(sc - msc) / esc;            // softmax(scores)
    lds_kl[tid] = tg * (__logf(tg) - __logf(p + EPSV));
    lds_pd[tid] = sc / __log2f((float)(rp + 2));        // gain / log2(rank+2)
    lds_id[tid] = sc / __log2f((float)(ri + 2));
  }
  __syncthreads();

  if (tid == 0) {  // deterministic sequential row reduction
    float skl = 0.f, spd = 0.f, sid = 0.f;
    for (int j = 0; j < An; ++j) { skl += lds_kl[j]; spd += lds_pd[j]; sid += lds_id[j]; }
    kl_out[b] = skl;
    nd_out[b] = spd / (sid + EPSV);
  }
}

// Deterministic final reduction: loss = mean(kl), avg_ndcg = mean(ratio).
__global__ __launch_bounds__(256) void final_reduce_kernel(
    const float* __restrict__ kl_row, const float* __restrict__ nd_row,
    float* __restrict__ out)
{
  __shared__ float skl[256];
  __shared__ float snd[256];
  const int t = threadIdx.x;
  float a = 0.f, n = 0.f;
  for (int i = t; i < Bn; i += 256) { a += kl_row[i]; n += nd_row[i]; }
  skl[t] = a; snd[t] = n;
  __syncthreads();
  for (int s = 128; s > 0; s >>= 1) {
    if (t < s) { skl[t] += skl[t + s]; snd[t] += snd[t + s]; }
    __syncthreads();
  }
  if (t == 0) {
    out[0] = skl[0] / (float)Bn;   // loss
    out[1] = snd[0] / (float)Bn;   // avg ndcg
  }
}

extern "C" void kernel_launch(void* const* d_in, const int* in_sizes, int n_in,
                              void* d_out, int out_size, void* d_ws, size_t ws_size,
                              hipStream_t stream) {
  const float* qemb   = (const float*)d_in[0];  // [1024,1024]
  const float* ans    = (const float*)d_in[1];  // [1024,64,1024]
  const float* scores = (const float*)d_in[2];  // [1024,64]
  float* kl_ws = (float*)d_ws;        // [1024]
  float* nd_ws = kl_ws + Bn;          // [1024]

  sim_rank_kernel<<<dim3(Bn), dim3(128), 0, stream>>>(qemb, ans, scores, kl_ws, nd_ws);
  final_reduce_kernel<<<dim3(1), dim3(256), 0, stream>>>(kl_ws, nd_ws, (float*)d_out);
}